// Gatv2CN_60009283060272
// MI455X (gfx1250) — compile-verified
//
#include <hip/hip_runtime.h>

#define N_NODES 50000
#define N_EDGES 800000
#define IN_DIM 64
#define HEADS 4
#define CH 32
#define DIM 128
#define EDIM 4
#define NGRAPH 2000
#define NTASK 8
#define EE (N_EDGES + N_NODES)
#define NEG_SLOPE 0.2f
#define EPS_SM 1e-16f

typedef __attribute__((ext_vector_type(2))) float v2f;
typedef __attribute__((ext_vector_type(8))) float v8f;
typedef int v4i_vs __attribute__((vector_size(16)));  // matches builtin param type

__device__ __forceinline__ float fatomic_add(float* p, float v) {
  // relaxed, agent scope -> hardware global_atomic_add_f32
  return __hip_atomic_fetch_add(p, v, __ATOMIC_RELAXED, __HIP_MEMORY_SCOPE_AGENT);
}

__device__ __forceinline__ void fatomic_max(float* p, float v) {
  // float max via signed/unsigned integer ordering trick (init must be -inf)
  if (v >= 0.0f) atomicMax((int*)p, __float_as_int(v));
  else           atomicMin((unsigned int*)p, __float_as_uint(v));
}

__global__ void k_zero(float* __restrict__ p, int n) {
  int i = blockIdx.x * blockDim.x + threadIdx.x;
  if (i < n) p[i] = 0.0f;
}

// column sums of edge_attr (E x 4) -> easum[4]
__global__ void k_easum(const float* __restrict__ ea, float* __restrict__ easum) {
  __shared__ float sh[256];
  float p[4] = {0.f, 0.f, 0.f, 0.f};
  for (int e = blockIdx.x * blockDim.x + threadIdx.x; e < N_EDGES;
       e += gridDim.x * blockDim.x) {
    float4 v = ((const float4*)ea)[e];
    p[0] += v.x; p[1] += v.y; p[2] += v.z; p[3] += v.w;
  }
  for (int k = 0; k < 4; ++k) {
    sh[threadIdx.x] = p[k];
    __syncthreads();
    for (int s = 128; s > 0; s >>= 1) {
      if ((int)threadIdx.x < s) sh[threadIdx.x] += sh[threadIdx.x + s];
      __syncthreads();
    }
    if (threadIdx.x == 0) fatomic_add(&easum[k], sh[0]);
    __syncthreads();
  }
}

// Y[N,128] = X[N,din] @ W[din,128] + bias.  din in {64,128}, N % 16 == 0.
// Block = 256 threads = 8 wave32; wave w owns output columns [16w, 16w+16).
// A tile (16 x din fp32, contiguous in X) staged in LDS via CDNA5 async
// copy (global_load_async_to_lds_b128 + s_wait_asynccnt) when available;
// K advanced 4 at a time through V_WMMA_F32_16X16X4_F32 (full fp32).
__global__ void k_gemm_wmma(const float* __restrict__ X, const float* __restrict__ W,
                            const float* __restrict__ bias, float* __restrict__ Y,
                            int din) {
  __shared__ __align__(16) float Alds[16 * 128];
  const int tid = threadIdx.x;
  const int m0 = blockIdx.x * 16;
  const float* Asrc = X + (size_t)m0 * din;  // 16-row tile is contiguous
  const int tile = 16 * din;
#if __has_builtin(__builtin_amdgcn_global_load_async_to_lds_b128)
  for (int f = tid * 4; f < tile; f += 256 * 4) {
    __builtin_amdgcn_global_load_async_to_lds_b128(
        (__attribute__((address_space(1))) v4i_vs*)(Asrc + f),
        (__attribute__((address_space(3))) v4i_vs*)(&Alds[f]),
        0, 0);
  }
#if __has_builtin(__builtin_amdgcn_s_wait_asynccnt)
  __builtin_amdgcn_s_wait_asynccnt(0);
#else
  asm volatile("s_wait_asynccnt 0x0" ::: "memory");
#endif
  __syncthreads();
#else
  for (int f = tid * 4; f < tile; f += 256 * 4) {
    *(float4*)(&Alds[f]) = *(const float4*)(Asrc + f);
  }
  __syncthreads();
#endif
  const int wave = tid >> 5;
  const int lane = tid & 31;
  const int lo = lane & 15;
  const int hi = lane >> 4;
  const int n0 = wave * 16;
  v8f acc = {};
  for (int k0 = 0; k0 < din; k0 += 4) {
    const int ka = k0 + 2 * hi;
    v2f a, b;
    // 16x4 f32 A layout: lanes 0-15 hold K=k0,k0+1; lanes 16-31 hold K=k0+2,k0+3
    a.x = Alds[lo * din + ka];
    a.y = Alds[lo * din + ka + 1];
    // 4x16 f32 B layout mirrors A with N striped across lanes
    b.x = W[ka * DIM + n0 + lo];
    b.y = W[(ka + 1) * DIM + n0 + lo];
    acc = __builtin_amdgcn_wmma_f32_16x16x4_f32(false, a, false, b,
                                                (short)0, acc, false, false);
  }
  const float bv = bias[n0 + lo];
#pragma unroll
  for (int r = 0; r < 8; ++r) {
    // C/D layout: VGPR r -> row m0 + r + 8*hi, col n0 + lo
    Y[(m0 + r + 8 * hi) * DIM + n0 + lo] = acc[r] + bv;
  }
}

// per-layer init: out = broadcast(bo), amax = -inf, den = 0
__global__ void k_init_layer(float* __restrict__ amax, float* __restrict__ den,
                             float* __restrict__ out, const float* __restrict__ bo) {
  int i = blockIdx.x * blockDim.x + threadIdx.x;
  if (i < N_NODES * DIM) out[i] = bo[i & (DIM - 1)];
  if (i < N_NODES * HEADS) {
    amax[i] = -__builtin_huge_valf();
    den[i] = 0.0f;
  }
}

// Pass 1: wave per edge, B128 gathers. Lane owns channels d = 4*lane..4*lane+3
// (all in head h = lane>>3).  alpha[e][h] = att[h] . leaky(xl[src]+xr[dst]+em)
// with em = ea[e] @ We computed on the fly (We is 2KB, cache resident).
// Head-wise reduction = 3-step shfl_xor within 8-lane groups; atomic
// segment-max into amax[dst][h] done by lanes 0/8/16/24.
__global__ void k_edge_alpha(const float* __restrict__ xl, const float* __restrict__ xr,
                             const int* __restrict__ ei, const float* __restrict__ eattr,
                             const float* __restrict__ easum,
                             const float* __restrict__ We, const float* __restrict__ att,
                             float* __restrict__ alpha, float* __restrict__ amax) {
  int gt = blockIdx.x * blockDim.x + threadIdx.x;
  int e = gt >> 5;
  int lane = gt & 31;
  if (e >= EE) return;  // wave-uniform
  int src, dst;
  float ea0, ea1, ea2, ea3;
  if (e < N_EDGES) {
    src = ei[e]; dst = ei[N_EDGES + e];
    float4 eav = ((const float4*)eattr)[e];
    ea0 = eav.x; ea1 = eav.y; ea2 = eav.z; ea3 = eav.w;
  } else {
    src = dst = e - N_EDGES;  // self loop, edge_attr = column mean
    const float inv = 1.0f / (float)N_EDGES;
    ea0 = easum[0] * inv; ea1 = easum[1] * inv;
    ea2 = easum[2] * inv; ea3 = easum[3] * inv;
  }
  const int h = lane >> 3;
  float4 xlv = ((const float4*)(xl + (size_t)src * DIM))[lane];
  float4 xrv = ((const float4*)(xr + (size_t)dst * DIM))[lane];
  float4 a4  = ((const float4*)att)[lane];
  float4 w0  = ((const float4*)(We + 0 * DIM))[lane];
  float4 w1  = ((const float4*)(We + 1 * DIM))[lane];
  float4 w2  = ((const float4*)(We + 2 * DIM))[lane];
  float4 w3  = ((const float4*)(We + 3 * DIM))[lane];
  float part = 0.0f, m;
  m = xlv.x + xrv.x + (ea0 * w0.x + ea1 * w1.x + ea2 * w2.x + ea3 * w3.x);
  part += a4.x * ((m > 0.0f) ? m : NEG_SLOPE * m);
  m = xlv.y + xrv.y + (ea0 * w0.y + ea1 * w1.y + ea2 * w2.y + ea3 * w3.y);
  part += a4.y * ((m > 0.0f) ? m : NEG_SLOPE * m);
  m = xlv.z + xrv.z + (ea0 * w0.z + ea1 * w1.z + ea2 * w2.z + ea3 * w3.z);
  part += a4.z * ((m > 0.0f) ? m : NEG_SLOPE * m);
  m = xlv.w + xrv.w + (ea0 * w0.w + ea1 * w1.w + ea2 * w2.w + ea3 * w3.w);
  part += a4.w * ((m > 0.0f) ? m : NEG_SLOPE * m);
  part += __shfl_xor(part, 4, 32);
  part += __shfl_xor(part, 2, 32);
  part += __shfl_xor(part, 1, 32);
  if ((lane & 7) == 0) {
    alpha[e * HEADS + h] = part;
    fatomic_max(&amax[dst * HEADS + h], part);
  }
}

// Pass 2: thread per (edge, head). ex = exp(alpha - amax[dst]); den += ex.
__global__ void k_edge_exp(const int* __restrict__ ei, float* __restrict__ alpha,
                           const float* __restrict__ amax, float* __restrict__ den) {
  int gid = blockIdx.x * blockDim.x + threadIdx.x;
  if (gid >= EE * HEADS) return;
  int e = gid >> 2, h = gid & 3;
  int dst = (e < N_EDGES) ? ei[N_EDGES + e] : (e - N_EDGES);
  float ex = __expf(alpha[gid] - amax[dst * HEADS + h]);
  alpha[gid] = ex;
  fatomic_add(&den[dst * HEADS + h], ex);
}

// Pass 3: out[dst] += xl[src] * ex/(den[dst]+eps).  B128 gather of xl,
// hardware f32 atomic adds into L2 for the scatter.
__global__ void k_edge_scatter(const float* __restrict__ xl, const int* __restrict__ ei,
                               const float* __restrict__ alpha, const float* __restrict__ den,
                               float* __restrict__ out) {
  int gt = blockIdx.x * blockDim.x + threadIdx.x;
  int e = gt >> 5;
  int lane = gt & 31;
  if (e >= EE) return;  // wave-uniform
  int src, dst;
  if (e < N_EDGES) { src = ei[e]; dst = ei[N_EDGES + e]; }
  else             { src = dst = e - N_EDGES; }
  const int h = lane >> 3;
  float w = alpha[e * HEADS + h] / (den[dst * HEADS + h] + EPS_SM);
  float4 xlv = ((const float4*)(xl + (size_t)src * DIM))[lane];
  float* op = out + (size_t)dst * DIM + lane * 4;
  fatomic_add(op + 0, xlv.x * w);
  fatomic_add(op + 1, xlv.y * w);
  fatomic_add(op + 2, xlv.z * w);
  fatomic_add(op + 3, xlv.w * w);
}

__global__ void k_relu(float* __restrict__ p, int n) {
  int i = blockIdx.x * blockDim.x + threadIdx.x;
  if (i < n) p[i] = fmaxf(p[i], 0.0f);
}

// global mean pool accumulation
__global__ void k_pool(const float* __restrict__ h, const int* __restrict__ batch,
                       float* __restrict__ xg, float* __restrict__ cnt) {
  int i = blockIdx.x * blockDim.x + threadIdx.x;
  if (i >= N_NODES * DIM) return;
  int n = i >> 7, d = i & 127;
  int g = batch[n];
  fatomic_add(&xg[g * DIM + d], h[i]);
  if (d == 0) fatomic_add(&cnt[g], 1.0f);
}

// out[g][t] = (xg[g]/max(cnt,1)) . Wt[t] + bt[t]
__global__ void k_heads(const float* __restrict__ xg, const float* __restrict__ cnt,
                        const float* __restrict__ Wt, const float* __restrict__ bt,
                        float* __restrict__ out) {
  int i = blockIdx.x * blockDim.x + threadIdx.x;
  if (i >= NGRAPH * NTASK) return;
  int g = i / NTASK, t = i - g * NTASK;
  float inv = 1.0f / fmaxf(cnt[g], 1.0f);
  float acc = 0.0f;
  for (int d = 0; d < DIM; ++d) acc += xg[g * DIM + d] * Wt[t * DIM + d];
  out[i] = acc * inv + bt[t];
}

extern "C" void kernel_launch(void* const* d_in, const int* in_sizes, int n_in,
                              void* d_out, int out_size, void* d_ws, size_t ws_size,
                              hipStream_t stream) {
  (void)in_sizes; (void)n_in; (void)out_size; (void)ws_size;
  const float* x     = (const float*)d_in[0];
  const int*   ei    = (const int*)d_in[1];
  const float* eattr = (const float*)d_in[2];
  const int*   batch = (const int*)d_in[3];
  // d_in[4] = r_target (truthy -> all heads); unused
  const float* Wl[3] = {(const float*)d_in[5],  (const float*)d_in[12], (const float*)d_in[19]};
  const float* bl[3] = {(const float*)d_in[6],  (const float*)d_in[13], (const float*)d_in[20]};
  const float* Wr[3] = {(const float*)d_in[7],  (const float*)d_in[14], (const float*)d_in[21]};
  const float* br[3] = {(const float*)d_in[8],  (const float*)d_in[15], (const float*)d_in[22]};
  const float* We[3] = {(const float*)d_in[9],  (const float*)d_in[16], (const float*)d_in[23]};
  const float* att[3]= {(const float*)d_in[10], (const float*)d_in[17], (const float*)d_in[24]};
  const float* bo[3] = {(const float*)d_in[11], (const float*)d_in[18], (const float*)d_in[25]};
  const float* Wt = (const float*)d_in[26];
  const float* bt = (const float*)d_in[27];
  float* out = (float*)d_out;

  float* ws = (float*)d_ws;
  size_t o = 0;
  float* xl    = ws + o; o += (size_t)N_NODES * DIM;
  float* xr    = ws + o; o += (size_t)N_NODES * DIM;
  float* hA    = ws + o; o += (size_t)N_NODES * DIM;
  float* hB    = ws + o; o += (size_t)N_NODES * DIM;
  float* alpha = ws + o; o += (size_t)EE * HEADS;
  float* amax  = ws + o; o += (size_t)N_NODES * HEADS;
  float* den   = ws + o; o += (size_t)N_NODES * HEADS;
  float* xg    = ws + o; o += (size_t)NGRAPH * DIM;
  float* cnt   = ws + o; o += (size_t)NGRAPH;
  float* easum = ws + o; o += 4;

  const int B = 256;

  // edge_attr column mean (layer-invariant)
  k_zero<<<1, 64, 0, stream>>>(easum, 4);
  k_easum<<<256, B, 0, stream>>>(eattr, easum);

  const float* lin[3]  = {x, hA, hB};
  float*       lout[3] = {hA, hB, hA};
  const int    dins[3] = {IN_DIM, DIM, DIM};

  for (int L = 0; L < 3; ++L) {
    k_gemm_wmma<<<N_NODES / 16, B, 0, stream>>>(lin[L], Wl[L], bl[L], xl, dins[L]);
    k_gemm_wmma<<<N_NODES / 16, B, 0, stream>>>(lin[L], Wr[L], br[L], xr, dins[L]);
    k_init_layer<<<(N_NODES * DIM + B - 1) / B, B, 0, stream>>>(amax, den, lout[L], bo[L]);
    k_edge_alpha<<<(int)(((size_t)EE * 32 + B - 1) / B), B, 0, stream>>>(
        xl, xr, ei, eattr, easum, We[L], att[L], alpha, amax);
    k_edge_exp<<<(EE * HEADS + B - 1) / B, B, 0, stream>>>(ei, alpha, amax, den);
    k_edge_scatter<<<(int)(((size_t)EE * 32 + B - 1) / B), B, 0, stream>>>(
        xl, ei, alpha, den, lout[L]);
    if (L < 2) k_relu<<<(N_NODES * DIM + B - 1) / B, B, 0, stream>>>(lout[L], N_NODES * DIM);
  }

  k_zero<<<(NGRAPH * DIM + B - 1) / B, B, 0, stream>>>(xg, NGRAPH * DIM);
  k_zero<<<(NGRAPH + B - 1) / B, B, 0, stream>>>(cnt, NGRAPH);
  k_pool<<<(N_NODES * DIM + B - 1) / B, B, 0, stream>>>(lout[2], batch, xg, cnt);
  k_heads<<<(NGRAPH * NTASK + B - 1) / B, B, 0, stream>>>(xg, cnt, Wt, bt, out);
}